// GCNConvLayer_32238024524457
// MI455X (gfx1250) — compile-verified
//
#include <hip/hip_runtime.h>
#include <hip/hip_bf16.h>

typedef __attribute__((ext_vector_type(2))) float v2f;
typedef __attribute__((ext_vector_type(8))) float v8f;

#define D 128          // feature dim (in == out == 128)
#define DEG_CLAMP 1e6f

// ---------------------------------------------------------------- kernel 1
// deg starts at 1.0 to account for the appended self-loop per node.
__global__ void k_init_deg(float* __restrict__ deg, int N) {
    int i = blockIdx.x * blockDim.x + threadIdx.x;
    if (i < N) deg[i] = 1.0f;
}

// ---------------------------------------------------------------- kernel 2
// degree of destination nodes (edge_index is int64 per reference)
__global__ void k_count_deg(const long long* __restrict__ ei, float* __restrict__ deg,
                            int E) {
    int e = blockIdx.x * blockDim.x + threadIdx.x;
    if (e < E) {
        long long d = ei[(long long)E + e];   // dst row of edge_index
        atomicAdd(&deg[d], 1.0f);
    }
}

// ---------------------------------------------------------------- kernel 3
__global__ void k_inv_sqrt(const float* __restrict__ deg, float* __restrict__ dis, int N) {
    int i = blockIdx.x * blockDim.x + threadIdx.x;
    if (i < N) {
        float v = __frsqrt_rn(deg[i]);
        dis[i] = fminf(v, DEG_CLAMP);
    }
}

// ---------------------------------------------------------------- kernel 4
// h = x @ W^T + b via V_WMMA_F32_16X16X4_F32.
// Block = 128 threads = 4 waves. Each wave computes a 32x32 tile of h:
// 2 row-tiles x 2 col-tiles = 4 independent accumulator chains; each A/B
// operand feeds two WMMAs (4 WMMA per 4 b64 loads per K-step).
__global__ __launch_bounds__(128) void k_gemm_wmma(
        const float* __restrict__ x,   // [N,128] row-major
        const float* __restrict__ W,   // [128,128] row-major (D_OUT x D_IN)
        const float* __restrict__ bias,// [128]
        float* __restrict__ h,         // [N,128]
        int N) {
    const int wave  = threadIdx.x >> 5;
    const int lane  = threadIdx.x & 31;
    const int l15   = lane & 15;
    const int khalf = lane >> 4;          // 0: K={0,1}, 1: K={2,3} within a K-block
    const int row0  = blockIdx.x * 32;
    const int col0  = wave * 32;

    // A-matrix rows (clamped so EXEC stays all-1s across the WMMAs)
    const int ar0 = min(row0      + l15, N - 1);
    const int ar1 = min(row0 + 16 + l15, N - 1);
    const float* xp0 = x + (size_t)ar0 * D + khalf * 2;
    const float* xp1 = x + (size_t)ar1 * D + khalf * 2;
    // B-matrix (= W^T): B[k][n] = W[n][k]; lane holds col n = colX + l15
    const float* wp0 = W + (size_t)(col0      + l15) * D + khalf * 2;
    const float* wp1 = W + (size_t)(col0 + 16 + l15) * D + khalf * 2;

    v8f c00 = {}, c01 = {}, c10 = {}, c11 = {};
#pragma unroll 4
    for (int kb = 0; kb < D; kb += 4) {
        v2f a0 = *(const v2f*)(xp0 + kb);
        v2f a1 = *(const v2f*)(xp1 + kb);
        v2f b0 = *(const v2f*)(wp0 + kb);
        v2f b1 = *(const v2f*)(wp1 + kb);
        c00 = __builtin_amdgcn_wmma_f32_16x16x4_f32(false, a0, false, b0,
                                                    (short)0, c00, false, false);
        c01 = __builtin_amdgcn_wmma_f32_16x16x4_f32(false, a0, false, b1,
                                                    (short)0, c01, false, false);
        c10 = __builtin_amdgcn_wmma_f32_16x16x4_f32(false, a1, false, b0,
                                                    (short)0, c10, false, false);
        c11 = __builtin_amdgcn_wmma_f32_16x16x4_f32(false, a1, false, b1,
                                                    (short)0, c11, false, false);
    }

    const float bias0 = bias[col0 + l15];
    const float bias1 = bias[col0 + 16 + l15];

    if (row0 + 32 <= N) {
        // fast path: whole tile in-bounds, no per-row predication
#pragma unroll
        for (int v = 0; v < 8; ++v) {
            int m = v + khalf * 8;               // C/D layout: VGPR v, lane half
            size_t r0 = (size_t)(row0      + m) * D;
            size_t r1 = (size_t)(row0 + 16 + m) * D;
            h[r0 + col0      + l15] = c00[v] + bias0;
            h[r0 + col0 + 16 + l15] = c01[v] + bias1;
            h[r1 + col0      + l15] = c10[v] + bias0;
            h[r1 + col0 + 16 + l15] = c11[v] + bias1;
        }
    } else {
#pragma unroll
        for (int v = 0; v < 8; ++v) {
            int m = v + khalf * 8;
            int row = row0 + m;
            if (row < N) {
                h[(size_t)row * D + col0      + l15] = c00[v] + bias0;
                h[(size_t)row * D + col0 + 16 + l15] = c01[v] + bias1;
            }
            if (row + 16 < N) {
                h[(size_t)(row + 16) * D + col0      + l15] = c10[v] + bias0;
                h[(size_t)(row + 16) * D + col0 + 16 + l15] = c11[v] + bias1;
            }
        }
    }
}

// ---------------------------------------------------------------- kernel 5
// Self-loop contribution initializes out: out[i,:] = h[i,:] * dis[i]^2.
// float4-vectorized: thread t handles features [4t, 4t+4) of node t/32.
__global__ void k_self_loop(const float4* __restrict__ h4, const float* __restrict__ dis,
                            float4* __restrict__ out4, long long total4) {
    long long idx = (long long)blockIdx.x * blockDim.x + threadIdx.x;
    if (idx < total4) {
        long long node = idx >> 5;               // 32 float4 per node
        float s = dis[node];
        float n = s * s;
        float4 v = h4[idx];
        v.x *= n; v.y *= n; v.z *= n; v.w *= n;
        out4[idx] = v;
    }
}

// ---------------------------------------------------------------- kernel 6
// One wave per edge; each lane handles 4 features (float4 gather + 4 atomics).
// Index/norm loads are wave-uniform -> compiler scalarizes them.
__global__ __launch_bounds__(256) void k_scatter(
        const long long* __restrict__ ei, const float* __restrict__ h,
        const float* __restrict__ dis, float* __restrict__ out, int E) {
    const int wave = threadIdx.x >> 5;
    const int lane = threadIdx.x & 31;
    long long e = (long long)blockIdx.x * 8 + wave;
    if (e >= E) return;
    long long s = ei[e];
    long long d = ei[(long long)E + e];
    float norm = dis[s] * dis[d];
    const float4 hv = *(const float4*)(h + s * (long long)D + lane * 4);
    float* o = out + d * (long long)D + lane * 4;
    atomicAdd(o + 0, hv.x * norm);
    atomicAdd(o + 1, hv.y * norm);
    atomicAdd(o + 2, hv.z * norm);
    atomicAdd(o + 3, hv.w * norm);
}

// ---------------------------------------------------------------- launcher
extern "C" void kernel_launch(void* const* d_in, const int* in_sizes, int n_in,
                              void* d_out, int out_size, void* d_ws, size_t ws_size,
                              hipStream_t stream) {
    const float*     x  = (const float*)d_in[0];      // [N,128]
    const long long* ei = (const long long*)d_in[1];  // [2,E] int64
    const float*     W  = (const float*)d_in[2];      // [128,128]
    const float*     b  = (const float*)d_in[3];      // [128]
    float*           out = (float*)d_out;             // [N,128]

    const int N = in_sizes[0] / D;
    const int E = in_sizes[1] / 2;

    // workspace layout: deg[N] | dis[N] | h[N*128]
    float* deg = (float*)d_ws;
    float* dis = deg + N;
    float* h   = dis + N;

    const int T = 256;
    k_init_deg <<<(N + T - 1) / T, T, 0, stream>>>(deg, N);
    k_count_deg<<<(E + T - 1) / T, T, 0, stream>>>(ei, deg, E);
    k_inv_sqrt <<<(N + T - 1) / T, T, 0, stream>>>(deg, dis, N);

    const int rowTiles = (N + 31) / 32;               // 32 rows per block
    k_gemm_wmma<<<rowTiles, 128, 0, stream>>>(x, W, b, h, N);

    long long total4 = (long long)N * (D / 4);
    k_self_loop<<<(int)((total4 + T - 1) / T), T, 0, stream>>>(
        (const float4*)h, dis, (float4*)out, total4);

    k_scatter<<<(E + 7) / 8, T, 0, stream>>>(ei, h, dis, out, E);
}